// My_GATLayer_20864951124309
// MI455X (gfx1250) — compile-verified
//
#include <hip/hip_runtime.h>
#include <hip/hip_bf16.h>

#define DFEAT 256
#define APITCH 260   // A-tile LDS row pitch in floats (256 + 4 pad) -> conflict-free ds reads

typedef float v2f __attribute__((ext_vector_type(2)));
typedef float v8f __attribute__((ext_vector_type(8)));
typedef unsigned int u32x4 __attribute__((ext_vector_type(4)));
typedef int i32x4 __attribute__((ext_vector_type(4)));
typedef int i32x8 __attribute__((ext_vector_type(8)));

// ---------------------------------------------------------------------------
// Kernel 1: fused dual GEMM.  hs = h @ Wself^T,  z = h @ Wfunc^T
// One block = 32 nodes (2 node-tiles). A tile (32x256 f32) DMA'd into LDS by
// the Tensor Data Mover with row padding (pitch 260) for conflict-free reads.
// 8 waves; each wave owns 2 col-tiles x 2 node-tiles x 2 outputs
// -> 8 x v8f accumulators, 8 x V_WMMA_F32_16X16X4_F32 per K-step.
// B loads are software-pipelined one K-step ahead.
// ---------------------------------------------------------------------------
__global__ __launch_bounds__(256) void gat_gemm_kernel(
    const float* __restrict__ h, const float* __restrict__ Wself,
    const float* __restrict__ Wfunc, float* __restrict__ hs,
    float* __restrict__ z)
{
    __shared__ float tileA[32 * APITCH];   // 33.28 KB
    const int node0 = blockIdx.x * 32;

    // ---- TDM: one tensor_load_to_lds for the whole 32x256 A tile ----------
    if (threadIdx.x < 32) {                // wave 0 issues the DMA
        const unsigned long long ga =
            (unsigned long long)(h + (size_t)node0 * DFEAT);
        const unsigned lds = (unsigned)(unsigned long long)(void*)tileA;

        u32x4 g0 = {};
        g0[0] = 1u;                                        // count=1, user D#
        g0[1] = lds;                                       // lds_addr (bytes)
        g0[2] = (unsigned)(ga & 0xffffffffull);            // global_addr[31:0]
        g0[3] = (unsigned)((ga >> 32) & 0x1ffffffull)      // global_addr[56:32]
              | (2u << 30);                                // type = 2 (image)

        i32x8 g1 = {};
        // data_size=4B (2<<16) | pad_enable (1<<20) | pad_interval=256 DW (7<<22)
        // | pad_amount=4 DW (3<<25)
        g1[0] = (2 << 16) | (1 << 20) | (7 << 22) | (3 << 25);
        g1[1] = (int)(256u << 16);      // tensor_dim0[15:0] = 256  (bits 79:48)
        g1[2] = (int)(32u  << 16);      // tensor_dim1[15:0] = 32   (bits 111:80)
        g1[3] = (int)(256u << 16);      // dim1 hi = 0 | tile_dim0 = 256
        g1[4] = 32;                     // tile_dim1 = 32 rows, tile_dim2 = 0
        g1[5] = 256;                    // tensor_dim0_stride = 256 elements
        g1[6] = 0; g1[7] = 0;           // tensor_dim1_stride = 0 (2-D)

        i32x4 gz4 = {};
        i32x8 gz8 = {};
        __builtin_amdgcn_tensor_load_to_lds(g0, g1, gz4, gz4, gz8, 0);
        __builtin_amdgcn_s_wait_tensorcnt(0);
    }
    __syncthreads();

    const int wave = threadIdx.x >> 5;
    const int lane = threadIdx.x & 31;
    const int m    = lane & 15;   // A row / B,C col within tile
    const int kh   = lane >> 4;   // k-half selector (K 0,1 vs 2,3)
    const int ct0  = wave * 2;    // first of this wave's two col-tiles

    v8f s00 = {}; v8f s01 = {}; v8f s10 = {}; v8f s11 = {};
    v8f z00 = {}; v8f z01 = {}; v8f z10 = {}; v8f z11 = {};

    // B[k][n] = W[(ct*16+n)*D + k]  (W row-major, row = output col)
    const float* pBs0 = Wself + (size_t)( ct0      * 16 + m) * DFEAT + kh * 2;
    const float* pBs1 = Wself + (size_t)((ct0 + 1) * 16 + m) * DFEAT + kh * 2;
    const float* pBf0 = Wfunc + (size_t)( ct0      * 16 + m) * DFEAT + kh * 2;
    const float* pBf1 = Wfunc + (size_t)((ct0 + 1) * 16 + m) * DFEAT + kh * 2;
    const float* Ar0  = tileA + (size_t)m        * APITCH + kh * 2;
    const float* Ar1  = tileA + (size_t)(m + 16) * APITCH + kh * 2;

    v2f bs0 = *(const v2f*)(pBs0);
    v2f bs1 = *(const v2f*)(pBs1);
    v2f bf0 = *(const v2f*)(pBf0);
    v2f bf1 = *(const v2f*)(pBf1);

    #pragma unroll 4
    for (int k0 = 0; k0 < DFEAT; k0 += 4) {
        // issue next K-step's B loads before this step's WMMAs (pipelining)
        v2f nbs0, nbs1, nbf0, nbf1;
        if (k0 + 4 < DFEAT) {
            nbs0 = *(const v2f*)(pBs0 + k0 + 4);
            nbs1 = *(const v2f*)(pBs1 + k0 + 4);
            nbf0 = *(const v2f*)(pBf0 + k0 + 4);
            nbf1 = *(const v2f*)(pBf1 + k0 + 4);
        }
        v2f a0 = *(const v2f*)(Ar0 + k0);
        v2f a1 = *(const v2f*)(Ar1 + k0);

        s00 = __builtin_amdgcn_wmma_f32_16x16x4_f32(false, a0, false, bs0, (short)0, s00, false, false);
        s10 = __builtin_amdgcn_wmma_f32_16x16x4_f32(false, a1, false, bs0, (short)0, s10, false, false);
        s01 = __builtin_amdgcn_wmma_f32_16x16x4_f32(false, a0, false, bs1, (short)0, s01, false, false);
        s11 = __builtin_amdgcn_wmma_f32_16x16x4_f32(false, a1, false, bs1, (short)0, s11, false, false);
        z00 = __builtin_amdgcn_wmma_f32_16x16x4_f32(false, a0, false, bf0, (short)0, z00, false, false);
        z10 = __builtin_amdgcn_wmma_f32_16x16x4_f32(false, a1, false, bf0, (short)0, z10, false, false);
        z01 = __builtin_amdgcn_wmma_f32_16x16x4_f32(false, a0, false, bf1, (short)0, z01, false, false);
        z11 = __builtin_amdgcn_wmma_f32_16x16x4_f32(false, a1, false, bf1, (short)0, z11, false, false);

        bs0 = nbs0; bs1 = nbs1; bf0 = nbf0; bf1 = nbf1;
    }

    // C layout: VGPR r, lanes 0-15 -> M=r, lanes 16-31 -> M=8+r; N = lane&15
    #pragma unroll
    for (int r = 0; r < 8; ++r) {
        const size_t row0 = (size_t)(node0 + kh * 8 + r);
        const size_t off0 = row0 * DFEAT + ct0 * 16 + m;
        const size_t off1 = (row0 + 16) * DFEAT + ct0 * 16 + m;
        hs[off0]      = s00[r];
        z [off0]      = z00[r];
        hs[off0 + 16] = s01[r];
        z [off0 + 16] = z01[r];
        hs[off1]      = s10[r];
        z [off1]      = z10[r];
        hs[off1 + 16] = s11[r];
        z [off1 + 16] = z11[r];
    }
}

// ---------------------------------------------------------------------------
// Kernel 2: per-node attention scalars  s[i] = z_i . Ws,  d[i] = z_i . Wd
// ---------------------------------------------------------------------------
__global__ __launch_bounds__(256) void gat_dots_kernel(
    const float* __restrict__ z, const float* __restrict__ Wattn,
    float* __restrict__ sA, float* __restrict__ dA, int N)
{
    const int i = blockIdx.x * 8 + (threadIdx.x >> 5);
    const int lane = threadIdx.x & 31;
    if (i >= N) return;
    float as = 0.f, ad = 0.f;
    const float* zi = z + (size_t)i * DFEAT;
    #pragma unroll
    for (int k = lane; k < DFEAT; k += 32) {
        float zv = zi[k];
        as += zv * Wattn[k];
        ad += zv * Wattn[DFEAT + k];
    }
    #pragma unroll
    for (int off = 16; off > 0; off >>= 1) {
        as += __shfl_xor(as, off, 32);
        ad += __shfl_xor(ad, off, 32);
    }
    if (lane == 0) { sA[i] = as; dA[i] = ad; }
}

// ---------------------------------------------------------------------------
// Kernel 3: init scratch (graph-replay safe: re-initialized every call)
// ---------------------------------------------------------------------------
__global__ void gat_init_kernel(float* __restrict__ agg, float* __restrict__ mA,
                                float* __restrict__ denom, float* __restrict__ deg,
                                float* __restrict__ g, long long ND, int N)
{
    long long idx = (long long)blockIdx.x * 256 + threadIdx.x;
    if (idx < ND) agg[idx] = 0.f;
    if (idx < N) { mA[idx] = -__builtin_inff(); denom[idx] = 0.f; deg[idx] = 0.f; }
    if (idx == 0) { g[0] = -__builtin_inff(); g[1] = 0.f; }
}

// ---------------------------------------------------------------------------
// Kernel 4: edge logits e = leaky_relu(s[src]+d[dst]); segment max (atomic
// float max -> GLOBAL_ATOMIC_MAX_NUM_F32) and global max (block-reduced)
// ---------------------------------------------------------------------------
__global__ __launch_bounds__(256) void gat_edge_e_kernel(
    const float* __restrict__ sA, const float* __restrict__ dA,
    const int* __restrict__ src, const int* __restrict__ dst,
    float* __restrict__ eA, float* __restrict__ mA, float* __restrict__ g, int E)
{
    const int idx = blockIdx.x * 256 + threadIdx.x;
    float ev = -__builtin_inff();
    if (idx < E) {
        float x = sA[src[idx]] + dA[dst[idx]];
        ev = (x > 0.f) ? x : 0.01f * x;        // jax leaky_relu slope 0.01
        eA[idx] = ev;
        atomicMax(&mA[dst[idx]], ev);
    }
    __shared__ float red[256];
    red[threadIdx.x] = ev;
    __syncthreads();
    for (int off = 128; off > 0; off >>= 1) {
        if (threadIdx.x < off) red[threadIdx.x] = fmaxf(red[threadIdx.x], red[threadIdx.x + off]);
        __syncthreads();
    }
    if (threadIdx.x == 0) atomicMax(&g[0], red[0]);
}

// ---------------------------------------------------------------------------
// Kernel 5: e_exp = exp(e - m[dst]); denom/deg segment sums; global exp-sum
// ---------------------------------------------------------------------------
__global__ __launch_bounds__(256) void gat_edge_exp_kernel(
    const float* __restrict__ eA, const int* __restrict__ dst,
    const float* __restrict__ mA, const float* __restrict__ g,
    float* __restrict__ eExp, float* __restrict__ denom,
    float* __restrict__ deg, float* __restrict__ gsum, int E)
{
    const int idx = blockIdx.x * 256 + threadIdx.x;
    float gv = 0.f;
    if (idx < E) {
        const int dN = dst[idx];
        const float ev = eA[idx];
        const float ee = expf(ev - mA[dN]);
        eExp[idx] = ee;
        atomicAdd(&denom[dN], ee);
        atomicAdd(&deg[dN], 1.0f);
        gv = expf(ev - g[0]);
    }
    __shared__ float red[256];
    red[threadIdx.x] = gv;
    __syncthreads();
    for (int off = 128; off > 0; off >>= 1) {
        if (threadIdx.x < off) red[threadIdx.x] += red[threadIdx.x + off];
        __syncthreads();
    }
    if (threadIdx.x == 0) atomicAdd(gsum, red[0]);
}

// ---------------------------------------------------------------------------
// Kernel 6: agg[dst] += alpha * z[src]  (one wave per edge, float atomics),
// plus the global-softmax output a_global
// ---------------------------------------------------------------------------
__global__ __launch_bounds__(256) void gat_edge_agg_kernel(
    const float* __restrict__ eA, const float* __restrict__ eExp,
    const float* __restrict__ denom, const float* __restrict__ z,
    const int* __restrict__ src, const int* __restrict__ dst,
    const float* __restrict__ g, float* __restrict__ agg,
    float* __restrict__ a_global, int E)
{
    const int edge = blockIdx.x * 8 + (threadIdx.x >> 5);
    const int lane = threadIdx.x & 31;
    if (edge >= E) return;
    const int sN = src[edge], dN = dst[edge];
    const float alpha = eExp[edge] / denom[dN];
    const float* zs = z + (size_t)sN * DFEAT;
    float* ag = agg + (size_t)dN * DFEAT;
    #pragma unroll
    for (int k = lane; k < DFEAT; k += 32)
        atomicAdd(&ag[k], alpha * zs[k]);
    if (lane == 0)
        a_global[edge] = expf(eA[edge] - g[0]) / g[1];
}

// ---------------------------------------------------------------------------
// Kernel 7: out = h + relu( deg>0 ? hs+agg : h )
// ---------------------------------------------------------------------------
__global__ __launch_bounds__(256) void gat_final_kernel(
    const float* __restrict__ h, const float* __restrict__ hs,
    const float* __restrict__ agg, const float* __restrict__ deg,
    float* __restrict__ out, long long ND)
{
    const long long idx = (long long)blockIdx.x * 256 + threadIdx.x;
    if (idx >= ND) return;
    const int i = (int)(idx / DFEAT);
    const float hin = h[idx];
    const float hmsg = (deg[i] > 0.f) ? (hs[idx] + agg[idx]) : hin;
    out[idx] = hin + fmaxf(hmsg, 0.f);
}

// ---------------------------------------------------------------------------
extern "C" void kernel_launch(void* const* d_in, const int* in_sizes, int n_in,
                              void* d_out, int out_size, void* d_ws, size_t ws_size,
                              hipStream_t stream)
{
    const float* h     = (const float*)d_in[0];
    // d_in[1] = snorm_n : unused by the reference computation
    const float* Wself = (const float*)d_in[2];
    const float* Wfunc = (const float*)d_in[3];
    const float* Wattn = (const float*)d_in[4];
    const int*   src   = (const int*)d_in[5];
    const int*   dst   = (const int*)d_in[6];

    const int N = in_sizes[1];
    const int E = in_sizes[5];
    const size_t ND = (size_t)N * DFEAT;

    // workspace layout (floats): hs | z | agg | s | d | m | denom | deg | e | e_exp | gmax,gsum
    float* ws    = (float*)d_ws;
    float* hs    = ws;
    float* z     = ws + ND;
    float* agg   = ws + 2 * ND;
    float* sA    = ws + 3 * ND;
    float* dA    = sA + N;
    float* mA    = dA + N;
    float* denom = mA + N;
    float* deg   = denom + N;
    float* eA    = deg + N;
    float* eExp  = eA + E;
    float* g     = eExp + E;  // g[0] = global max, g[1] = global exp-sum

    float* out      = (float*)d_out;
    float* a_global = out + ND;

    const int ndBlocks = (int)((ND + 255) / 256);

    gat_init_kernel<<<ndBlocks, 256, 0, stream>>>(agg, mA, denom, deg, g, (long long)ND, N);
    gat_gemm_kernel<<<N / 32, 256, 0, stream>>>(h, Wself, Wfunc, hs, z);
    gat_dots_kernel<<<(N + 7) / 8, 256, 0, stream>>>(z, Wattn, sA, dA, N);
    gat_edge_e_kernel<<<(E + 255) / 256, 256, 0, stream>>>(sA, dA, src, dst, eA, mA, g, E);
    gat_edge_exp_kernel<<<(E + 255) / 256, 256, 0, stream>>>(eA, dst, mA, g, eExp, denom, deg, g + 1, E);
    gat_edge_agg_kernel<<<(E + 7) / 8, 256, 0, stream>>>(eA, eExp, denom, z, src, dst, g, agg, a_global, E);
    gat_final_kernel<<<ndBlocks, 256, 0, stream>>>(h, hs, agg, deg, out, (long long)ND);
}